// Model_8022998909551
// MI455X (gfx1250) — compile-verified
//
#include <hip/hip_runtime.h>
#include <hip/hip_bf16.h>
#include <math.h>

#define T_DIM   720
#define FEATD   512
#define NND     1024
#define D_DIM   8192      // 2 * 64 * 64
#define N2      16384     // 2 * D
#define S2      4096      // 64 * 64
#define LOG2PI  1.8378770664093453f
#define LN10    2.302585092994046f
#define LN10000 9.210340371976184f

typedef float v2f __attribute__((ext_vector_type(2)));
typedef float v8f __attribute__((ext_vector_type(8)));

__device__ __forceinline__ v8f wmma4(v2f a, v2f b, v8f c) {
  // V_WMMA_F32_16X16X4_F32 : D = A(16x4,f32) * B(4x16,f32) + C(16x16,f32)
  return __builtin_amdgcn_wmma_f32_16x16x4_f32(false, a, false, b, (short)0, c,
                                               false, false);
}

__device__ __forceinline__ float sigm(float x) { return 1.f / (1.f + expf(-x)); }

// ---------------------------------------------------------------------------
// K1: h = sigmoid(features @ W1 + b1)   (720x512)@(512x1024)
// One wave -> 16(M) x 64(N) tile. grid = (45, 16), block = 32.
// ---------------------------------------------------------------------------
extern "C" __global__ void __launch_bounds__(32)
k_gemm1(const float* __restrict__ A, const float* __restrict__ W,
        const float* __restrict__ bias, float* __restrict__ H) {
  const int lane = threadIdx.x;
  const int m0 = blockIdx.x * 16;
  const int n0 = blockIdx.y * 64;
  const int r  = lane & 15;    // A row / B-D column (per ISA VGPR layout)
  const int kh = lane >> 4;    // K half-select
  const float* Arow = A + (size_t)(m0 + r) * FEATD;
  v8f acc[4] = {};
  for (int k0 = 0; k0 < FEATD; k0 += 4) {
    const int ka = k0 + 2 * kh;
    v2f a; a.x = Arow[ka]; a.y = Arow[ka + 1];
    const float* B0 = W + (size_t)ka * NND + n0 + r;
    const float* B1 = B0 + NND;
#pragma unroll
    for (int tN = 0; tN < 4; ++tN) {
      v2f b; b.x = B0[tN * 16]; b.y = B1[tN * 16];
      acc[tN] = wmma4(a, b, acc[tN]);
    }
  }
#pragma unroll
  for (int tN = 0; tN < 4; ++tN) {
#pragma unroll
    for (int i = 0; i < 8; ++i) {
      const int mm = m0 + i + 8 * kh;   // C/D layout: VGPR i holds M = i + 8*kh
      const int nn = n0 + tN * 16 + r;
      H[(size_t)mm * NND + nn] = sigm(acc[tN][i] + bias[nn]);
    }
  }
}

// ---------------------------------------------------------------------------
// K2: loc_scale = h @ W2 + b2 ; epilogue emits r = y - loc  and rinv = 1/sc^2
// (720x1024)@(1024x16384). grid = (45, 256), block = 32.
// ---------------------------------------------------------------------------
extern "C" __global__ void __launch_bounds__(32)
k_gemm2(const float* __restrict__ H, const float* __restrict__ W,
        const float* __restrict__ bias, const float* __restrict__ Y,
        float* __restrict__ Rv, float* __restrict__ Rinv) {
  const int lane = threadIdx.x;
  const int m0 = blockIdx.x * 16;
  const int n0 = blockIdx.y * 64;
  const int r  = lane & 15;
  const int kh = lane >> 4;
  const float* Arow = H + (size_t)(m0 + r) * NND;
  v8f acc[4] = {};
  for (int k0 = 0; k0 < NND; k0 += 4) {
    const int ka = k0 + 2 * kh;
    v2f a; a.x = Arow[ka]; a.y = Arow[ka + 1];
    const float* B0 = W + (size_t)ka * N2 + n0 + r;
    const float* B1 = B0 + N2;
    __builtin_prefetch(B0 + 16 * (size_t)N2, 0, 0);   // global_prefetch_b8
#pragma unroll
    for (int tN = 0; tN < 4; ++tN) {
      v2f b; b.x = B0[tN * 16]; b.y = B1[tN * 16];
      acc[tN] = wmma4(a, b, acc[tN]);
    }
  }
#pragma unroll
  for (int tN = 0; tN < 4; ++tN) {
#pragma unroll
    for (int i = 0; i < 8; ++i) {
      const int mm = m0 + i + 8 * kh;
      const int nn = n0 + tN * 16 + r;
      float v = acc[tN][i] + bias[nn];
      if (nn < D_DIM) {
        // loc column: r = y - loc
        Rv[(size_t)mm * D_DIM + nn] = Y[(size_t)mm * D_DIM + nn] - v;
      } else {
        // scale column: sc = bounded_exp(v, 10) ; store 1/sc^2
        float sc = 10.f * sigm(v - LN10);
        Rinv[(size_t)mm * D_DIM + (nn - D_DIM)] = 1.f / (sc * sc);
      }
    }
  }
}

// ---------------------------------------------------------------------------
// K3: per-t stats: rr = sum(rinv*r^2), slog = sum(log sc),
//     crr[16] = (C*rinv)@r  (VALU + tree reduce)
//     G[16][16] = C diag(rinv) C^T   (WMMA f32 16x16x4, K split over 8 waves)
// rinv[t] is staged into LDS once via CDNA5 async DMA (ASYNCcnt path) and
// consumed by both phases.
// One block (256 thr) per t. stats layout per t (stride 288):
//   [0..255]=G, [256..271]=crr, [272]=rr, [273]=slog
// ---------------------------------------------------------------------------
extern "C" __global__ void __launch_bounds__(256)
k_stats(const float* __restrict__ Rv, const float* __restrict__ Rinv,
        const float* __restrict__ C, float* __restrict__ stats) {
  const int t = blockIdx.x, tid = threadIdx.x;
  const float* rv = Rv + (size_t)t * D_DIM;
  const float* ri = Rinv + (size_t)t * D_DIM;

  __shared__ float ri_l[D_DIM];      // 32 KB: staged rinv[t]
  __shared__ float red[256][18];     // 18 KB: scalar reductions
  __shared__ float gpart[8][256];    //  8 KB: per-wave partial Gram matrices

  // --- stage rinv[t] -> LDS with global_load_async_to_lds_b128 (ASYNCcnt) ---
  {
    const unsigned lds0 = (unsigned)(size_t)&ri_l[0];
#pragma unroll
    for (int c = 0; c < 8; ++c) {
      const int fidx = (c * 256 + tid) * 4;          // 4 floats per b128 op
      const unsigned ldsa = lds0 + (unsigned)fidx * 4u;
      const unsigned voff = (unsigned)fidx * 4u;     // byte offset vs saddr
      asm volatile("global_load_async_to_lds_b128 %0, %1, %2"
                   :: "v"(ldsa), "v"(voff), "s"(ri)
                   : "memory");
    }
    asm volatile("s_wait_asynccnt 0" ::: "memory");
  }
  __syncthreads();

  // --- phase A: rr, slog, crr ---
  float rr = 0.f, slog = 0.f, crr[16];
#pragma unroll
  for (int i = 0; i < 16; ++i) crr[i] = 0.f;
  for (int d = tid; d < D_DIM; d += 256) {
    float r = rv[d], q = ri_l[d];
    rr += q * r * r;
    slog += -0.5f * logf(q);          // log(sc) = -0.5*log(rinv)
    float qr = q * r;
#pragma unroll
    for (int i = 0; i < 16; ++i) crr[i] += C[i * D_DIM + d] * qr;
  }
  red[tid][0] = rr; red[tid][1] = slog;
#pragma unroll
  for (int i = 0; i < 16; ++i) red[tid][2 + i] = crr[i];
  __syncthreads();
  for (int s = 128; s > 0; s >>= 1) {     // deterministic tree reduction
    if (tid < s)
      for (int v = 0; v < 18; ++v) red[tid][v] += red[tid + s][v];
    __syncthreads();
  }

  // --- phase B: G = C diag(rinv) C^T via WMMA, K=8192 split over 8 waves ---
  const int w    = tid >> 5;
  const int lane = tid & 31;
  const int r16  = lane & 15;
  const int kh   = lane >> 4;
  const float* Crow = C + (size_t)r16 * D_DIM;
  v8f acc = {};
  const int kbeg = w * (D_DIM / 8);
  for (int k0 = kbeg; k0 < kbeg + D_DIM / 8; k0 += 4) {
    const int ka = k0 + 2 * kh;                 // even -> 8B aligned
    v2f c2 = *(const v2f*)(Crow + ka);          // C[r16][ka..ka+1]
    v2f q2 = *(const v2f*)(&ri_l[ka]);          // rinv from LDS (ds_load_b64)
    v2f a = c2 * q2;                            // (C * rinv) row  -> A
    acc = wmma4(a, c2, acc);                    // B column n = r16 is same C row
  }
#pragma unroll
  for (int i = 0; i < 8; ++i)
    gpart[w][(i + 8 * kh) * 16 + r16] = acc[i]; // D layout: M = i + 8*kh, N = r16
  __syncthreads();

  float* st = stats + (size_t)t * 288;
  {
    float g = 0.f;
#pragma unroll
    for (int w2 = 0; w2 < 8; ++w2) g += gpart[w2][tid];
    st[tid] = g;
  }
  if (tid < 16) st[256 + tid] = red[0][2 + tid];
  if (tid == 0) { st[272] = red[0][0]; st[273] = red[0][1]; }
}

// ---------------------------------------------------------------------------
// K4: sequential Kalman scan (720 steps of 16x16 algebra). 1 block of 32.
// ---------------------------------------------------------------------------
__device__ void chol16(float (*S)[17], int tid) {
  for (int k = 0; k < 16; ++k) {
    if (tid == k) S[k][k] = sqrtf(S[k][k]);
    __syncthreads();
    if (tid < 16 && tid > k) S[tid][k] = S[tid][k] / S[k][k];
    __syncthreads();
    if (tid < 16 && tid > k) {
      float lik = S[tid][k];
      for (int j = k + 1; j <= tid; ++j) S[tid][j] -= lik * S[j][k];
    }
    __syncthreads();
  }
}

__device__ void chosolve_eye(const float (*L)[17], float (*X)[17], int tid) {
  if (tid < 16) {                 // one RHS column per lane
    const int c = tid;
    float y[16], x[16];
#pragma unroll
    for (int i2 = 0; i2 < 16; ++i2) {
      float s = (i2 == c) ? 1.f : 0.f;
#pragma unroll
      for (int k2 = 0; k2 < 16; ++k2)
        if (k2 < i2) s -= L[i2][k2] * y[k2];
      y[i2] = s / L[i2][i2];
    }
#pragma unroll
    for (int i2 = 15; i2 >= 0; --i2) {
      float s = y[i2];
#pragma unroll
      for (int k2 = 0; k2 < 16; ++k2)
        if (k2 > i2) s -= L[k2][i2] * x[k2];
      x[i2] = s / L[i2][i2];
    }
#pragma unroll
    for (int i2 = 0; i2 < 16; ++i2) X[i2][c] = x[i2];
  }
  __syncthreads();
}

extern "C" __global__ void __launch_bounds__(32)
k_scan(const float* __restrict__ stats, const float* __restrict__ Atm,
       const float* __restrict__ tscale, const float* __restrict__ iscale,
       float* __restrict__ out) {
  __shared__ float Am[16][16], P[16][17], Tm[16][17], Pp[16][17],
                   Pi[16][17], Gm[16][17], Lam[16][17], Pn[16][17];
  __shared__ float qd[16], mvec[16], mp[16], crr[16], uvec[16], eta[16],
                   redA[16], redB[16], scal[4];
  const int tid = threadIdx.x;
  for (int idx = tid; idx < 256; idx += 32) {
    int i = idx >> 4, j = idx & 15;
    Am[i][j] = Atm[idx];
    P[i][j] = (i == j) ? iscale[i] * iscale[i] : 0.f;
  }
  if (tid < 16) { mvec[tid] = 0.f; qd[tid] = tscale[tid] * tscale[tid]; }
  __syncthreads();
  float ll = 0.f;
  for (int t = 0; t < T_DIM; ++t) {
    const float* st = stats + (size_t)t * 288;
    for (int idx = tid; idx < 256; idx += 32) Gm[idx >> 4][idx & 15] = st[idx];
    if (tid < 16) crr[tid] = st[256 + tid];
    if (tid == 0) { scal[0] = st[272]; scal[1] = st[273]; }
    __syncthreads();
    // m_pred = A m ; Tm = A @ P
    if (tid < 16) {
      float s = 0.f;
      for (int j = 0; j < 16; ++j) s += Am[tid][j] * mvec[j];
      mp[tid] = s;
    }
    for (int idx = tid; idx < 256; idx += 32) {
      int i = idx >> 4, j = idx & 15;
      float s = 0.f;
      for (int k = 0; k < 16; ++k) s += Am[i][k] * P[k][j];
      Tm[i][j] = s;
    }
    __syncthreads();
    // P_pred = Tm @ A^T + Q
    for (int idx = tid; idx < 256; idx += 32) {
      int i = idx >> 4, j = idx & 15;
      float s = (i == j) ? qd[i] : 0.f;
      for (int k = 0; k < 16; ++k) s += Tm[i][k] * Am[j][k];
      Pp[i][j] = s;
    }
    __syncthreads();
    chol16(Pp, tid);
    if (tid == 0) {
      float s = 0.f;
      for (int k = 0; k < 16; ++k) s += logf(Pp[k][k]);
      scal[2] = s;
    }
    chosolve_eye(Pp, Pi, tid);                       // P_inv
    for (int idx = tid; idx < 256; idx += 32) {      // Lam = P_inv + G
      int i = idx >> 4, j = idx & 15;
      Lam[i][j] = Pi[i][j] + Gm[i][j];
    }
    __syncthreads();
    chol16(Lam, tid);
    if (tid == 0) {
      float s = 0.f;
      for (int k = 0; k < 16; ++k) s += logf(Lam[k][k]);
      scal[3] = s;
    }
    chosolve_eye(Lam, Pn, tid);                      // P_new = Lam^{-1}
    if (tid < 16) {
      float s = 0.f;
      for (int j = 0; j < 16; ++j) s += Gm[tid][j] * mp[j];   // (G m_pred)_i
      uvec[tid] = crr[tid] - s;                               // u = crr - G m
      redA[tid] = mp[tid] * s - 2.f * mp[tid] * crr[tid];     // mGm - 2 m.crr
    }
    __syncthreads();
    if (tid < 16) {
      float s = 0.f;
      for (int j = 0; j < 16; ++j) s += Pn[tid][j] * uvec[j]; // Lam^{-1} u
      redB[tid] = uvec[tid] * s;
      float e = 0.f;
      for (int j = 0; j < 16; ++j) e += Pi[tid][j] * mp[j];
      eta[tid] = e + crr[tid];                                // P_inv m + crr
    }
    __syncthreads();
    if (tid == 0) {
      float quad = scal[0];                                   // rr
      for (int i = 0; i < 16; ++i) quad += redA[i] - redB[i];
      float logdet = 2.f * scal[1] + 2.f * scal[2] + 2.f * scal[3];
      ll += -0.5f * ((float)D_DIM * LOG2PI + logdet + quad);
    }
    if (tid < 16) {                                           // m_new = Pn eta
      float s = 0.f;
      for (int j = 0; j < 16; ++j) s += Pn[tid][j] * eta[j];
      mp[tid] = s;
    }
    __syncthreads();
    if (tid < 16) mvec[tid] = mp[tid];
    for (int idx = tid; idx < 256; idx += 32) {
      int i = idx >> 4, j = idx & 15;
      P[i][j] = Pn[i][j];
    }
    __syncthreads();
  }
  if (tid == 0) out[0] = ll;
}

// ---------------------------------------------------------------------------
// K5: zero-inflated Poisson likelihood, per-t block partial sums.
// ---------------------------------------------------------------------------
extern "C" __global__ void __launch_bounds__(256)
k_pois(const float* __restrict__ gate_rate, const int* __restrict__ counts,
       float* __restrict__ partial) {
  const int t = blockIdx.x, tid = threadIdx.x;
  const float* g = gate_rate + (size_t)t * D_DIM;
  const int* cn = counts + (size_t)t * S2;
  float acc = 0.f;
  for (int s = tid; s < S2; s += 256) {
    float gate  = g[s];
    float ratep = g[S2 + s];
    int   c     = cn[s];
    float rate = 10000.f * (1.f / (1.f + expf(-(ratep - LN10000))));
    float lse  = fmaxf(gate, 0.f) + log1pf(expf(-fabsf(gate)));  // logaddexp(0,gate)
    float lc0  = -lse;
    float lc1  = gate - lse;
    float cf   = (float)c;
    float pois = cf * logf(rate) - rate - lgammaf(cf + 1.f);
    float a    = lc0 + pois;
    float val;
    if (c == 0) {
      float mx = fmaxf(a, lc1);
      val = mx + log1pf(expf(-fabsf(a - lc1)));                  // logaddexp
    } else {
      val = a;
    }
    acc += val;
  }
  __shared__ float red[256];
  red[tid] = acc;
  __syncthreads();
  for (int s = 128; s > 0; s >>= 1) {
    if (tid < s) red[tid] += red[tid + s];
    __syncthreads();
  }
  if (tid == 0) partial[t] = red[0];
}

extern "C" __global__ void __launch_bounds__(256)
k_final(const float* __restrict__ partial, float* __restrict__ out) {
  __shared__ float red[256];
  const int tid = threadIdx.x;
  float a = 0.f;
  for (int i = tid; i < T_DIM; i += 256) a += partial[i];
  red[tid] = a;
  __syncthreads();
  for (int s = 128; s > 0; s >>= 1) {
    if (tid < s) red[tid] += red[tid + s];
    __syncthreads();
  }
  if (tid == 0) out[1] = red[0];
}

// ---------------------------------------------------------------------------
extern "C" void kernel_launch(void* const* d_in, const int* in_sizes, int n_in,
                              void* d_out, int out_size, void* d_ws,
                              size_t ws_size, hipStream_t stream) {
  (void)in_sizes; (void)n_in; (void)out_size; (void)ws_size;
  const float* features    = (const float*)d_in[0];
  const float* gate_rate   = (const float*)d_in[1];
  const float* W1          = (const float*)d_in[2];
  const float* b1          = (const float*)d_in[3];
  const float* W2          = (const float*)d_in[4];
  const float* b2          = (const float*)d_in[5];
  const float* init_scale  = (const float*)d_in[6];
  const float* trans_mat   = (const float*)d_in[7];
  const float* trans_scale = (const float*)d_in[8];
  const float* obs_matrix  = (const float*)d_in[9];
  const int*   trip_counts = (const int*)d_in[10];
  float* out = (float*)d_out;

  float* ws    = (float*)d_ws;
  float* h     = ws;                                // 720*1024
  float* rv    = h    + (size_t)T_DIM * NND;        // 720*8192
  float* rinv  = rv   + (size_t)T_DIM * D_DIM;      // 720*8192
  float* stats = rinv + (size_t)T_DIM * D_DIM;      // 720*288
  float* part  = stats + (size_t)T_DIM * 288;       // 720

  k_gemm1<<<dim3(T_DIM / 16, NND / 64), 32, 0, stream>>>(features, W1, b1, h);
  k_gemm2<<<dim3(T_DIM / 16, N2 / 64), 32, 0, stream>>>(h, W2, b2, gate_rate,
                                                        rv, rinv);
  k_stats<<<T_DIM, 256, 0, stream>>>(rv, rinv, obs_matrix, stats);
  k_scan<<<1, 32, 0, stream>>>(stats, trans_mat, trans_scale, init_scale, out);
  k_pois<<<T_DIM, 256, 0, stream>>>(gate_rate, trip_counts, part);
  k_final<<<1, 256, 0, stream>>>(part, out);
}